// SinDY_7687991460297
// MI455X (gfx1250) — compile-verified
//
#include <hip/hip_runtime.h>
#include <hip/hip_bf16.h>

// SINDy library regression:  out[b,t,:] = theta(x[b,t,:]) @ (mask * coeff)
//   x: (64, 8192, 16) f32   coeff/mask: (169, 16) f32   out: (64, 8192, 16) f32
//
// Memory-bound problem (~67 MB traffic -> ~2.9us at 23.3 TB/s); GEMM is only
// ~2.8 GFLOP so we use the f32 WMMA (V_WMMA_F32_16X16X4_F32) to keep the
// per-row instruction cost minimal while staying bit-faithful to the f32
// reference. K = 169 padded to 172 (43 WMMA steps of K=4).

#define LATENT 16
#define LIB 169              // 1 + 16 + 136 + 16
#define KPAD 172             // next multiple of 4 (pad features with 0)
#define THREADS 64           // 2 wave32 per block
#define ROWS_PER_BLOCK 64    // one row per thread in phase 1; 32 rows per wave in GEMM

typedef __attribute__((ext_vector_type(2))) float v2f;
typedef __attribute__((ext_vector_type(8))) float v8f;

// Feature k of the SINDy library; k is a compile-time constant after full
// unrolling, so the while-loop pair decoding and array indexing constant-fold
// and xr[] stays in registers.
__device__ __forceinline__ float sindy_feat(const float* xr, int k) {
  if (k == 0) return 1.0f;                         // constant term
  if (k <= 16) return xr[k - 1];                   // linear terms
  if (k <= 152) {                                  // 136 quadratic terms (a<=b, lexicographic)
    int t = k - 17;
    int a = 0;
    while (t >= LATENT - a) { t -= LATENT - a; ++a; }
    return xr[a] * xr[a + t];
  }
  if (k <= 168) return __sinf(xr[k - 153]);        // sin terms
  return 0.0f;                                     // zero padding (169..171)
}

__global__ __launch_bounds__(THREADS) void sindy_wmma_kernel(
    const float* __restrict__ x,
    const float* __restrict__ coef,
    const float* __restrict__ mask,
    float* __restrict__ out) {
  // Row stride KPAD=172 floats = 4*43 (43 odd): per-lane b64 reads below touch
  // all 64 LDS banks exactly once -> conflict-free.
  __shared__ float sTheta[ROWS_PER_BLOCK * KPAD];  // 44032 B
  __shared__ float sCmT[LATENT * KPAD];            // 11008 B (transposed, padded)

  const int tid = threadIdx.x;
  const size_t blockRow = (size_t)blockIdx.x * ROWS_PER_BLOCK;

  // ---- Stage B = (mask * coeff)^T into LDS, zero-padded to KPAD ----------
  // 16*172 = 2752 = 43 * THREADS, so the loop is divergence-free.
  for (int idx = tid; idx < LATENT * KPAD; idx += THREADS) {
    const int n = idx / KPAD;
    const int k = idx - n * KPAD;
    float v = 0.0f;
    if (k < LIB) v = coef[k * LATENT + n] * mask[k * LATENT + n];
    sCmT[idx] = v;
  }

  // ---- Phase 1: each lane builds theta for its own row in registers ------
  {
    const size_t rowg = blockRow + (size_t)tid;
    const float4* xp = reinterpret_cast<const float4*>(x + rowg * LATENT);
    float4 q0 = xp[0], q1 = xp[1], q2 = xp[2], q3 = xp[3];
    float xr[LATENT] = {q0.x, q0.y, q0.z, q0.w, q1.x, q1.y, q1.z, q1.w,
                        q2.x, q2.y, q2.z, q2.w, q3.x, q3.y, q3.z, q3.w};
    float* trow = &sTheta[tid * KPAD];
#pragma unroll
    for (int g = 0; g < KPAD / 4; ++g) {           // 43 x ds_store_b128
      float4 v;
      v.x = sindy_feat(xr, 4 * g + 0);
      v.y = sindy_feat(xr, 4 * g + 1);
      v.z = sindy_feat(xr, 4 * g + 2);
      v.w = sindy_feat(xr, 4 * g + 3);
      *reinterpret_cast<float4*>(trow + 4 * g) = v;
    }
  }

  __syncthreads();

  // ---- Phase 2: GEMM tiles via V_WMMA_F32_16X16X4_F32 --------------------
  // ISA fragment layouts (32-bit):
  //   A 16x4 : lanes 0-15 hold M=lane, K={kb,kb+1}; lanes 16-31 K={kb+2,kb+3}
  //   B 4x16 : VGPR0 lanes 0-15 -> K=kb row (N=lane), lanes 16-31 -> K=kb+2
  //   C/D    : VGPR i, lanes 0-15 -> M=i, lanes 16-31 -> M=i+8, N=lane%16
  const int wave = tid >> 5;
  const int lane = tid & 31;
  const int hi = lane >> 4;          // lane half
  const int n = lane & 15;           // A row index == B/D column index
  const int koff = hi * 2;

  const float* aBase0 = &sTheta[(wave * 32 + n) * KPAD + koff];
  const float* aBase1 = &sTheta[(wave * 32 + 16 + n) * KPAD + koff];
  const float* bBase  = &sCmT[n * KPAD + koff];

  v8f c0 = {0.f, 0.f, 0.f, 0.f, 0.f, 0.f, 0.f, 0.f};
  v8f c1 = {0.f, 0.f, 0.f, 0.f, 0.f, 0.f, 0.f, 0.f};
#pragma unroll
  for (int kb = 0; kb < KPAD; kb += 4) {           // 43 steps, 86 WMMAs/wave
    v2f a0 = *reinterpret_cast<const v2f*>(aBase0 + kb);
    v2f a1 = *reinterpret_cast<const v2f*>(aBase1 + kb);
    v2f b  = *reinterpret_cast<const v2f*>(bBase + kb);
    c0 = __builtin_amdgcn_wmma_f32_16x16x4_f32(false, a0, false, b,
                                               (short)0, c0, false, false);
    c1 = __builtin_amdgcn_wmma_f32_16x16x4_f32(false, a1, false, b,
                                               (short)0, c1, false, false);
  }

  // ---- Store D tiles (coalesced 64B per lane-half per component) ---------
  const size_t r0 = blockRow + (size_t)(wave * 32 + hi * 8);
#pragma unroll
  for (int i = 0; i < 8; ++i) {
    out[(r0 + i) * LATENT + n]      = c0[i];
    out[(r0 + 16 + i) * LATENT + n] = c1[i];
  }
}

extern "C" void kernel_launch(void* const* d_in, const int* in_sizes, int n_in,
                              void* d_out, int out_size, void* d_ws, size_t ws_size,
                              hipStream_t stream) {
  const float* x    = (const float*)d_in[0];  // (64, 8192, 16) f32
  const float* coef = (const float*)d_in[1];  // (169, 16) f32
  const float* mask = (const float*)d_in[2];  // (169, 16) f32
  float* out = (float*)d_out;                 // (64, 8192, 16) f32

  const int nrows = in_sizes[0] / LATENT;     // 524288, divisible by 64
  const int blocks = nrows / ROWS_PER_BLOCK;  // 8192
  sindy_wmma_kernel<<<blocks, THREADS, 0, stream>>>(x, coef, mask, out);
}